// NeighborlistVerletNsq_76115410419921
// MI455X (gfx1250) — compile-verified
//
#include <hip/hip_runtime.h>

// CDNA5 / gfx1250 neighbor-list N^2 kernel.
// Tiled 16x16 over the upper triangle; V_WMMA_F32_16X16X4_F32 computes the
// Gram cross-term r_i . r_j for 256 pairs per instruction; exact minimum-image
// correction applied per lane. Output layout (floats), P = N(N-1)/2:
//   [0,4P)    pairs rows: i | j | j | i          (indices as float)
//   [4P,6P)   d (masked), doubled
//   [6P,12P)  r (masked) [2P,3]: +r then -r
//   [12P,14P) mask as 1.0/0.0, doubled

typedef __attribute__((ext_vector_type(2))) float v2f;
typedef __attribute__((ext_vector_type(8))) float v8f;

#define TILE 16
#define CUTOFF_C 0.5f

__global__ __launch_bounds__(32)
void nsq_wmma_tile_kernel(const float* __restrict__ pos,
                          const float* __restrict__ box,   // 3x3 row-major, diag = box lengths
                          float* __restrict__ out,
                          int N, unsigned int P)
{
    const int bi = blockIdx.x;
    const int bj = blockIdx.y;
    if (bj < bi) return;                      // uniform: whole wave exits, EXEC stays full

    const int lane = threadIdx.x;             // 0..31 (wave32)
    const int half = lane >> 4;               // 0: lanes 0-15, 1: lanes 16-31
    const int col  = lane & 15;

    const int i0 = bi * TILE;
    const int j0 = bj * TILE;

    __shared__ float sI[TILE][3];
    __shared__ float sJ[TILE][3];
    __shared__ float sQI[TILE];
    __shared__ float sQJ[TILE];

    // ---- stage position blocks into LDS; lanes 0-15 -> I rows, 16-31 -> J rows
    {
        const int row  = col;
        int gidx = half ? (j0 + row) : (i0 + row);
        if (gidx > N - 1) gidx = N - 1;       // clamp (N is a multiple of 16 here)
        const float* src = pos + 3 * (size_t)gidx;
        const float x = src[0], y = src[1], z = src[2];
        const float q = x * x + y * y + z * z;
        if (half) { sJ[row][0] = x; sJ[row][1] = y; sJ[row][2] = z; sQJ[row] = q; }
        else      { sI[row][0] = x; sI[row][1] = y; sI[row][2] = z; sQI[row] = q; }
    }
    __syncthreads();

    // ---- WMMA fragments per ISA 16x4 f32 A-layout / 4x16 B-layout:
    //  A: vgpr0 = A[M=col][K = half?2:0], vgpr1 = A[M=col][K = half?3:1]
    //  B: vgpr0 = B[K = half?2:0][N=col], vgpr1 = B[K = half?3:1][N=col]
    v2f a, b;
    a.x = half ? sI[col][2] : sI[col][0];
    a.y = half ? 0.0f       : sI[col][1];
    b.x = half ? sJ[col][2] : sJ[col][0];
    b.y = half ? 0.0f       : sJ[col][1];

    v8f acc = {};
    // D = A x B + 0 : acc[r] (lane) = dot(pos[i0 + r + 8*half], pos[j0 + col])
    acc = __builtin_amdgcn_wmma_f32_16x16x4_f32(
        /*neg_a=*/false, a, /*neg_b=*/false, b,
        /*c_mod=*/(short)0, acc, /*reuse_a=*/false, /*reuse_b=*/false);

    // ---- box parameters (orthorhombic, diagonal box_vectors)
    const float Lx = box[0], Ly = box[4], Lz = box[8];
    const float iLx = 1.0f / Lx, iLy = 1.0f / Ly, iLz = 1.0f / Lz;

    const unsigned int n  = (unsigned int)col;
    const unsigned int j  = (unsigned int)(j0 + (int)n);
    const float pjx = sJ[n][0], pjy = sJ[n][1], pjz = sJ[n][2];
    const float qj  = sQJ[n];

    const size_t Ps = (size_t)P;

#pragma unroll
    for (int r = 0; r < 8; ++r) {
        const int m = r + 8 * half;                 // C/D layout: vgpr r <-> row m
        const unsigned int i = (unsigned int)(i0 + m);
        if (j <= i || j >= (unsigned int)N) continue;

        const float pix = sI[m][0], piy = sI[m][1], piz = sI[m][2];
        const float qi  = sQI[m];

        // raw displacement and minimum-image wrap counts
        const float rx = pix - pjx, ry = piy - pjy, rz = piz - pjz;
        const float nx = floorf(rx * iLx + 0.5f);
        const float ny = floorf(ry * iLy + 0.5f);
        const float nz = floorf(rz * iLz + 0.5f);
        const float wx = rx - Lx * nx;
        const float wy = ry - Ly * ny;
        const float wz = rz - Lz * nz;

        // exact ||w||^2 from the WMMA Gram term + integer-shift correction:
        // ||w||^2 = (qi + qj - 2*dot) - 2*sum(L_k n_k raw_k) + sum(L_k^2 n_k^2)
        const float dot = acc[r];
        float d2 = qi + qj - 2.0f * dot
                 - 2.0f * (Lx * nx * rx + Ly * ny * ry + Lz * nz * rz)
                 + (Lx * Lx * nx * nx + Ly * Ly * ny * ny + Lz * Lz * nz * nz);
        d2 = fmaxf(d2, 0.0f);
        const float d = sqrtf(d2);

        const bool keep = (d <= CUTOFF_C);
        const float dm  = keep ? d : 0.0f;
        const float mx  = keep ? wx : 0.0f;
        const float my  = keep ? wy : 0.0f;
        const float mz  = keep ? wz : 0.0f;
        const float mk  = keep ? 1.0f : 0.0f;

        // triangular linear index of (i, j) with i < j
        const size_t p = ((size_t)i * (2u * (unsigned int)N - 1u - i)) / 2u
                       + (size_t)(j - i - 1u);

        const float fi = (float)i, fj = (float)j;

        // pairs [2, 2P]: row0 = [i..., j...], row1 = [j..., i...]
        out[p]            = fi;
        out[Ps + p]       = fj;
        out[2 * Ps + p]   = fj;
        out[3 * Ps + p]   = fi;
        // d [2P,1]
        out[4 * Ps + p]   = dm;
        out[5 * Ps + p]   = dm;
        // r [2P,3]
        {
            const size_t rb = 6 * Ps + 3 * p;
            out[rb + 0] =  mx; out[rb + 1] =  my; out[rb + 2] =  mz;
            const size_t rn = 9 * Ps + 3 * p;     // 6P + 3*(P + p)
            out[rn + 0] = -mx; out[rn + 1] = -my; out[rn + 2] = -mz;
        }
        // mask [2P]
        out[12 * Ps + p]  = mk;
        out[13 * Ps + p]  = mk;
    }
}

extern "C" void kernel_launch(void* const* d_in, const int* in_sizes, int n_in,
                              void* d_out, int out_size, void* d_ws, size_t ws_size,
                              hipStream_t stream) {
    (void)n_in; (void)out_size; (void)d_ws; (void)ws_size;
    const float* pos = (const float*)d_in[0];   // [N,3]
    const float* box = (const float*)d_in[1];   // [3,3]
    const int N = in_sizes[0] / 3;
    const unsigned int P = (unsigned int)in_sizes[2];   // i_pairs length = N(N-1)/2

    const int NB = (N + TILE - 1) / TILE;       // 256 for N=4096
    dim3 grid((unsigned)NB, (unsigned)NB, 1);
    dim3 block(32, 1, 1);                       // one wave32 per 16x16 tile
    nsq_wmma_tile_kernel<<<grid, block, 0, stream>>>(pos, box, (float*)d_out, N, P);
}